// FConv2D_56788057588481
// MI455X (gfx1250) — compile-verified
//
#include <hip/hip_runtime.h>
#include <hip/hip_bf16.h>

// FNO-style spectral conv for MI455X (gfx1250), all DFTs done as GEMMs on the
// f32 WMMA pipe (V_WMMA_F32_16X16X4_F32). Shapes: B=16,H=256,W=256,C=D=64,
// Q=129 rfft bins (padded to 144 in storage, 132 in the irfft K-loop).
// Waves own full m-strips (all 4 c/d tiles) so A-matrix (twiddle) fragments
// are loaded once per K-step and reused across 8-16 WMMAs.
//
// Workspace layout (floats):
//   yre : [16][256][144][64]   (37,748,736)
//   yim : [16][256][144][64]   (37,748,736)
//   Fr,Fi : [144][256]         (36,864 each)   row-rfft twiddles (incl. 1/256)
//   Er,Ei : [256][256]         (65,536 each)   column-FFT twiddles
//   Cr,Ci : [256][132]         (33,792 each)   c2r irfft weights (incl. x256)
// total = 75,769,856 floats = ~303 MB  (ws_size must cover this)

typedef float v2f __attribute__((ext_vector_type(2)));
typedef float v8f __attribute__((ext_vector_type(8)));

static __device__ __forceinline__ v8f wmma4(v2f a, v2f b, v8f c) {
  // D(16x16,f32) = A(16x4,f32) * B(4x16,f32) + C
  return __builtin_amdgcn_wmma_f32_16x16x4_f32(
      false, a, false, b, (short)0, c, false, false);
}

#define NB 16
#define NH 256
#define NW 256
#define NC 64
#define QP 144   // padded q rows in storage (multiple of 16)
#define QD 132   // q rows consumed by irfft (multiple of 4, >=129)
#define NQ 129

// ---------------- twiddle tables -----------------------------------------

__global__ void init_F(float* __restrict__ Fr, float* __restrict__ Fi) {
  int idx = blockIdx.x * blockDim.x + threadIdx.x;
  if (idx >= QP * NW) return;
  int q = idx >> 8, w = idx & 255;
  float vr = 0.f, vi = 0.f;
  if (q < NQ) {
    int ph = (q * w) & 255;                       // exact mod-256 phase
    float ang = (float)ph * (3.14159265358979323846f / 128.f);
    float s, c; sincosf(ang, &s, &c);
    vr = c * (1.f / 256.f);                       // forward: e^{-i a}, /scale
    vi = -s * (1.f / 256.f);
  }
  Fr[idx] = vr; Fi[idx] = vi;
}

__global__ void init_E(float* __restrict__ Er, float* __restrict__ Ei) {
  int idx = blockIdx.x * blockDim.x + threadIdx.x;
  if (idx >= NH * NH) return;
  int t = idx >> 8, h = idx & 255;
  int ph = (t * h) & 255;
  float ang = (float)ph * (3.14159265358979323846f / 128.f);
  float s, c; sincosf(ang, &s, &c);
  Er[idx] = c; Ei[idx] = -s;                      // forward FFT over h
}

__global__ void init_C(float* __restrict__ Cr, float* __restrict__ Ci) {
  int idx = blockIdx.x * blockDim.x + threadIdx.x;
  if (idx >= NW * QD) return;
  int w = idx / QD, q = idx % QD;
  float vr = 0.f, vi = 0.f;
  if (q < NQ) {
    float wq = (q == 0 || q == 128) ? 1.f : 2.f;  // Hermitian mirror weight
    int ph = (q * w) & 255;
    float ang = (float)ph * (3.14159265358979323846f / 128.f);
    float s, c; sincosf(ang, &s, &c);
    vr = wq * c;                                  // (1/N from irfft) * scale
    vi = -wq * s;
  }
  Cr[idx] = vr; Ci[idx] = vi;
}

// ---------------- stage A: per-(b,h) row rFFT  [144x256]x[256x64] ---------
// 9 waves, one per q-tile strip; A frags shared across the 4 c-tiles.

__global__ __launch_bounds__(288) void stageA(
    const float* __restrict__ x, float* __restrict__ yre,
    float* __restrict__ yim, const float* __restrict__ Fr,
    const float* __restrict__ Fi) {
  extern __shared__ float xs[];                   // [256w][64c] = 64 KB
  int bh = blockIdx.x;
  const float* xrow = x + (size_t)bh * (NW * NC);
  for (int i = threadIdx.x; i < (NW * NC) / 4; i += blockDim.x)
    ((float4*)xs)[i] = ((const float4*)xrow)[i];
  __syncthreads();

  int wv = threadIdx.x >> 5, lane = threadIdx.x & 31;
  int sub = lane >> 4, li = lane & 15;
  size_t ybase = (size_t)bh * (QP * NC);

  int mt = wv;                                    // 9 waves <-> 9 q-tiles
  v8f accR[4] = {}, accI[4] = {};
  const float* frRow = Fr + (size_t)(mt * 16 + li) * NW;
  const float* fiRow = Fi + (size_t)(mt * 16 + li) * NW;
  for (int kk = 0; kk < NW; kk += 4) {
    int k = kk + 2 * sub;
    v2f ar = *(const v2f*)(frRow + k);
    v2f ai = *(const v2f*)(fiRow + k);
#pragma unroll
    for (int nt = 0; nt < 4; ++nt) {
      int n = nt * 16 + li;
      v2f b; b.x = xs[k * NC + n]; b.y = xs[(k + 1) * NC + n];
      accR[nt] = wmma4(ar, b, accR[nt]);
      accI[nt] = wmma4(ai, b, accI[nt]);
    }
  }
#pragma unroll
  for (int nt = 0; nt < 4; ++nt) {
    int n = nt * 16 + li;
#pragma unroll
    for (int r = 0; r < 8; ++r) {
      int q = mt * 16 + r + 8 * sub;
      yre[ybase + (size_t)q * NC + n] = accR[nt][r];
      yim[ybase + (size_t)q * NC + n] = accI[nt][r];
    }
  }
}

// ---------------- stage B: per-(b,q) column FFT over h, complex, in place --
// 8 waves x 2 t-tile strips; Er/Ei frags shared across 4 c-tiles (16 wmma/step)

__global__ __launch_bounds__(256) void stageB(
    float* __restrict__ yre, float* __restrict__ yim,
    const float* __restrict__ Er, const float* __restrict__ Ei) {
  extern __shared__ float sm[];                   // 2 x [256h][64c] = 128 KB
  float* yr_s = sm;
  float* yi_s = sm + NH * NC;
  int blk = blockIdx.x;
  int b = blk / NQ, q = blk % NQ;
  size_t base = (size_t)b * NH * (QP * NC) + (size_t)q * NC;

  for (int h = threadIdx.x; h < NH; h += blockDim.x) {
    const float4* sr = (const float4*)(yre + base + (size_t)h * (QP * NC));
    const float4* si = (const float4*)(yim + base + (size_t)h * (QP * NC));
    float4* dr = (float4*)(yr_s + h * NC);
    float4* di = (float4*)(yi_s + h * NC);
    for (int j = 0; j < NC / 4; ++j) { dr[j] = sr[j]; di[j] = si[j]; }
  }
  __syncthreads();

  int wv = threadIdx.x >> 5, lane = threadIdx.x & 31;
  int sub = lane >> 4, li = lane & 15;

  for (int mt = wv; mt < 16; mt += 8) {           // 2 strips per wave
    v8f accR[4] = {}, accI[4] = {};
    const float* erRow = Er + (size_t)(mt * 16 + li) * NH;
    const float* eiRow = Ei + (size_t)(mt * 16 + li) * NH;
    for (int kk = 0; kk < NH; kk += 4) {
      int k = kk + 2 * sub;
      v2f aer = *(const v2f*)(erRow + k);
      v2f aei = *(const v2f*)(eiRow + k);
      v2f aein = -aei;
#pragma unroll
      for (int nt = 0; nt < 4; ++nt) {
        int n = nt * 16 + li;
        v2f br; br.x = yr_s[k * NC + n]; br.y = yr_s[(k + 1) * NC + n];
        v2f bi; bi.x = yi_s[k * NC + n]; bi.y = yi_s[(k + 1) * NC + n];
        accR[nt] = wmma4(aer, br, accR[nt]);      // Zr = Er*Yr - Ei*Yi
        accR[nt] = wmma4(aein, bi, accR[nt]);
        accI[nt] = wmma4(aer, bi, accI[nt]);      // Zi = Er*Yi + Ei*Yr
        accI[nt] = wmma4(aei, br, accI[nt]);
      }
    }
#pragma unroll
    for (int nt = 0; nt < 4; ++nt) {
      int n = nt * 16 + li;
#pragma unroll
      for (int r = 0; r < 8; ++r) {
        int t = mt * 16 + r + 8 * sub;
        yre[base + (size_t)t * (QP * NC) + n] = accR[nt][r];
        yim[base + (size_t)t * (QP * NC) + n] = accI[nt][r];
      }
    }
  }
}

// ---------------- stage C: mode-block channel mixing  [16b x 64c]x[64c x 64d]

__global__ __launch_bounds__(128) void stageC(
    float* __restrict__ yre, float* __restrict__ yim,
    const float* __restrict__ w1, const float* __restrict__ w2) {
  extern __shared__ float sm[];                   // 2 x [16b][64c] = 8 KB
  float* zr = sm;
  float* zi = sm + 16 * NC;
  int id = blockIdx.x;
  int region = id >> 10, rem = id & 1023;
  int tp = rem >> 5, qq = rem & 31;
  int t = region ? (224 + tp) : tp;
  const float* Wb = region ? w2 : w1;             // [d][c][32][32][2]

  for (int i = threadIdx.x; i < 16 * NC; i += blockDim.x) {
    int bb = i >> 6, c = i & 63;
    size_t g = ((size_t)bb * NH + t) * (QP * NC) + (size_t)qq * NC + c;
    zr[i] = yre[g]; zi[i] = yim[g];
  }
  __syncthreads();

  int wv = threadIdx.x >> 5, lane = threadIdx.x & 31;
  int sub = lane >> 4, li = lane & 15;
  int nt = wv;                                    // 4 waves -> 4 d-tiles
  int d = nt * 16 + li;
  v8f accR = {}, accI = {};
  for (int kk = 0; kk < NC; kk += 4) {
    int k = kk + 2 * sub;
    v2f azr = *(const v2f*)(zr + li * NC + k);    // A[m=b][k=c]
    v2f azi = *(const v2f*)(zi + li * NC + k);
    v2f azin = -azi;
    size_t o0 = ((((size_t)d * 64 + k) * 32 + tp) * 32 + qq) * 2;
    size_t o1 = ((((size_t)d * 64 + k + 1) * 32 + tp) * 32 + qq) * 2;
    v2f p0 = *(const v2f*)(Wb + o0);              // {Wr,Wi} at c=k
    v2f p1 = *(const v2f*)(Wb + o1);              // {Wr,Wi} at c=k+1
    v2f bwr; bwr.x = p0.x; bwr.y = p1.x;
    v2f bwi; bwi.x = p0.y; bwi.y = p1.y;
    accR = wmma4(azr, bwr, accR);                 // Zr*Wr - Zi*Wi
    accR = wmma4(azin, bwi, accR);
    accI = wmma4(azr, bwi, accI);                 // Zr*Wi + Zi*Wr
    accI = wmma4(azi, bwr, accI);
  }
#pragma unroll
  for (int r = 0; r < 8; ++r) {
    int bb = r + 8 * sub;
    size_t g = ((size_t)bb * NH + t) * (QP * NC) + (size_t)qq * NC + d;
    yre[g] = accR[r];
    yim[g] = accI[r];
  }
}

// ---------------- stage D: per-(b,t) c2r irfft  [256w x 132q]x[132q x 64c] -
// 8 waves x 2 w-tile strips; Cr/Ci frags shared across 4 c-tiles.

__global__ __launch_bounds__(256) void stageD(
    const float* __restrict__ yre, const float* __restrict__ yim,
    const float* __restrict__ Cr, const float* __restrict__ Ci,
    float* __restrict__ out) {
  extern __shared__ float sm[];                   // 2 x [132q][64c] = 66 KB
  float* zr = sm;
  float* zi = sm + QD * NC;
  int bh = blockIdx.x;
  size_t base = (size_t)bh * (QP * NC);
  for (int i = threadIdx.x; i < (QD * NC) / 4; i += blockDim.x) {
    ((float4*)zr)[i] = ((const float4*)(yre + base))[i];
    ((float4*)zi)[i] = ((const float4*)(yim + base))[i];
  }
  __syncthreads();

  int wv = threadIdx.x >> 5, lane = threadIdx.x & 31;
  int sub = lane >> 4, li = lane & 15;
  size_t obase = (size_t)bh * (NW * NC);

  for (int mt = wv; mt < 16; mt += 8) {           // 2 strips per wave
    v8f acc[4] = {};
    const float* crRow = Cr + (size_t)(mt * 16 + li) * QD;
    const float* ciRow = Ci + (size_t)(mt * 16 + li) * QD;
    for (int kk = 0; kk < QD; kk += 4) {
      int k = kk + 2 * sub;
      v2f ac = *(const v2f*)(crRow + k);
      v2f as = *(const v2f*)(ciRow + k);
#pragma unroll
      for (int nt = 0; nt < 4; ++nt) {
        int n = nt * 16 + li;
        v2f br; br.x = zr[k * NC + n]; br.y = zr[(k + 1) * NC + n];
        v2f bi; bi.x = zi[k * NC + n]; bi.y = zi[(k + 1) * NC + n];
        acc[nt] = wmma4(ac, br, acc[nt]);         // + Cr*Zr
        acc[nt] = wmma4(as, bi, acc[nt]);         // + Ci*Zi
      }
    }
#pragma unroll
    for (int nt = 0; nt < 4; ++nt) {
      int n = nt * 16 + li;
#pragma unroll
      for (int r = 0; r < 8; ++r) {
        int w = mt * 16 + r + 8 * sub;
        out[obase + (size_t)w * NC + n] = acc[nt][r];
      }
    }
  }
}

// ---------------- launcher -------------------------------------------------

extern "C" void kernel_launch(void* const* d_in, const int* in_sizes, int n_in,
                              void* d_out, int out_size, void* d_ws,
                              size_t ws_size, hipStream_t stream) {
  (void)in_sizes; (void)n_in; (void)out_size; (void)ws_size;
  const float* x  = (const float*)d_in[0];
  const float* w1 = (const float*)d_in[1];
  const float* w2 = (const float*)d_in[2];
  float* out = (float*)d_out;
  float* ws  = (float*)d_ws;

  const size_t YSZ = (size_t)NB * NH * QP * NC;   // 37,748,736
  float* yre = ws;
  float* yim = ws + YSZ;
  float* Fr  = ws + 2 * YSZ;
  float* Fi  = Fr + QP * NW;
  float* Er  = Fi + QP * NW;
  float* Ei  = Er + NH * NH;
  float* Cr  = Ei + NH * NH;
  float* Ci  = Cr + NW * QD;

  init_F<<<(QP * NW + 255) / 256, 256, 0, stream>>>(Fr, Fi);
  init_E<<<(NH * NH + 255) / 256, 256, 0, stream>>>(Er, Ei);
  init_C<<<(NW * QD + 255) / 256, 256, 0, stream>>>(Cr, Ci);

  stageA<<<NB * NH, 288, NW * NC * 4, stream>>>(x, yre, yim, Fr, Fi);
  stageB<<<NB * NQ, 256, 2 * NH * NC * 4, stream>>>(yre, yim, Er, Ei);
  stageC<<<2 * 32 * 32, 128, 2 * 16 * NC * 4, stream>>>(yre, yim, w1, w2);
  stageD<<<NB * NH, 256, 2 * QD * NC * 4, stream>>>(yre, yim, Cr, Ci, out);
}